// RegionProposalNetwork_68702296866819
// MI455X (gfx1250) — compile-verified
//
#include <hip/hip_runtime.h>
#include <hip/hip_bf16.h>
#include <math.h>

// ---------------- problem constants ----------------
constexpr int B = 8, A = 3, H = 336, W = 336;
constexpr int HW = H * W;
constexpr int N = A * H * W;          // 338688 logits per image
constexpr int PRE = 2000;             // pre-NMS top-k
constexpr int POST = 1000;            // NMS output rows
constexpr float IMGF = 1344.0f;       // H * STRIDE
constexpr float BBOX_CLIP = 4.135166556742356f;   // ln(1000/16)
constexpr float NMS_THR = 0.7f;

// ordered key for -0.5f (validity threshold in NMS): bits(-0.5)=0xBF000000, neg -> ^0xFFFFFFFF
constexpr unsigned ORD_NEG_HALF = 0x40FFFFFFu;

// ---------------- CDNA5 async global->LDS path (guarded) ----------------
#if defined(__HIP_DEVICE_COMPILE__) && \
    __has_builtin(__builtin_amdgcn_global_load_async_to_lds_b128) && \
    __has_builtin(__builtin_amdgcn_global_load_async_to_lds_b32) && \
    __has_builtin(__builtin_amdgcn_s_wait_asynccnt)
#define RPN_USE_ASYNC_LDS 1
typedef int v4i __attribute__((ext_vector_type(4)));
typedef __attribute__((address_space(1))) v4i* gv4p;   // global int4*
typedef __attribute__((address_space(3))) v4i* lv4p;   // LDS int4*
typedef __attribute__((address_space(1))) int* gi1p;   // global int*
typedef __attribute__((address_space(3))) int* li1p;   // LDS int*
#endif

// monotonic float -> uint key (preserves total order)
__device__ __forceinline__ unsigned fkey(float f) {
  unsigned u = __float_as_uint(f);
  return u ^ (((unsigned)((int)u >> 31)) | 0x80000000u);
}
// inverse of fkey
__device__ __forceinline__ float funkey(unsigned u) {
  return __uint_as_float(u ^ ((u & 0x80000000u) ? 0x80000000u : 0xFFFFFFFFu));
}

// block-wide (1024 thr, 32 waves of 32) exclusive scan of a boolean, via wave32 ballot
__device__ __forceinline__ void block_scan_bool(bool p, unsigned* wsum,
                                                unsigned& excl, unsigned& total) {
  const unsigned lane = threadIdx.x & 31u;
  const unsigned wid  = threadIdx.x >> 5;
  const unsigned m = (unsigned)__ballot(p);                 // wave32: low 32 bits
  const unsigned pre = __popc(m & ((1u << lane) - 1u));
  if (lane == 0) wsum[wid] = __popc(m);
  __syncthreads();
  if (wid == 0) {
    unsigned v = wsum[lane];
    for (int off = 1; off < 32; off <<= 1) {
      unsigned u = __shfl_up(v, off);
      if (lane >= (unsigned)off) v += u;
    }
    wsum[lane] = v;                                          // inclusive scan
  }
  __syncthreads();
  excl  = ((wid == 0) ? 0u : wsum[wid - 1]) + pre;
  total = wsum[31];
  __syncthreads();                                           // wsum reusable after return
}

// ============================================================================
// Kernel 1: exact top-2000 per image via 4-pass MSD radix select + ordered
// compaction. One block (1024 threads) per image; data stays L2-resident.
// ============================================================================
__global__ __launch_bounds__(1024)
void rpn_topk_kernel(const float* __restrict__ obj,
                     int* __restrict__ sel_idx, float* __restrict__ sel_logit) {
  const int b = blockIdx.x;
  const int t = threadIdx.x;
  const float* __restrict__ src = obj + (size_t)b * (size_t)N;   // raw (A,H,W) order

  __shared__ unsigned hist[256];
  __shared__ unsigned wsum[32];
  __shared__ unsigned sh_prefix, sh_rem, sh_eqbase, sh_outbase;

  if (t == 0) { sh_prefix = 0u; sh_rem = (unsigned)PRE; }
  __syncthreads();

  // ---- phase A: find the exact key of the 2000th-largest element ----
  for (int p = 0; p < 4; ++p) {
    for (int i = t; i < 256; i += 1024) hist[i] = 0u;
    __syncthreads();
    const int shift = 24 - 8 * p;
    const unsigned hi_mask = (p == 0) ? 0u : (0xFFFFFFFFu << (shift + 8));
    const unsigned pref = sh_prefix;
    for (int j = t; j < N; j += 1024) {
      const unsigned k = fkey(src[j]);
      if ((k & hi_mask) == (pref & hi_mask))
        atomicAdd(&hist[(k >> shift) & 255u], 1u);
    }
    __syncthreads();
    if (t == 0) {
      unsigned rem = sh_rem, cum = 0;
      int d = 255;
      while (d > 0 && cum + hist[d] < rem) { cum += hist[d]; --d; }
      sh_rem = rem - cum;
      sh_prefix = pref | ((unsigned)d << shift);
    }
    __syncthreads();
  }
  const unsigned T = sh_prefix;          // exact ordered key of rank-2000 value
  const unsigned q = sh_rem;             // how many elements == T to take

  if (t == 0) { sh_eqbase = 0u; sh_outbase = 0u; }
  __syncthreads();

  // ---- phase B: deterministic ordered compaction of exactly PRE elements ----
  for (int chunk = 0; chunk < N; chunk += 1024) {
    const int j = chunk + t;
    const bool inb = (j < N);
    float f = 0.0f; unsigned k = 0u;
    if (inb) { f = src[j]; k = fkey(f); }
    const bool eq = inb && (k == T);
    const bool gt = inb && (k > T);

    unsigned eqx, eqtot;
    block_scan_bool(eq, wsum, eqx, eqtot);
    const bool take = gt || (eq && (sh_eqbase + eqx < q));

    unsigned tx, ttot;
    block_scan_bool(take, wsum, tx, ttot);

    if (take) {
      const unsigned pos = sh_outbase + tx;
      const int a = j / HW;
      const int hw = j - a * HW;
      sel_idx  [(size_t)b * PRE + pos] = hw * A + a;   // logical (H,W,A) flat index
      sel_logit[(size_t)b * PRE + pos] = f;
    }
    __syncthreads();
    if (t == 0) { sh_eqbase += eqtot; sh_outbase += ttot; }
    __syncthreads();
  }
}

// ============================================================================
// Kernel 2: gather deltas + anchors for the 2000 winners, decode, clip,
// sigmoid, min-size filter. Fully parallel (16000 threads).
// ============================================================================
__global__ __launch_bounds__(256)
void rpn_decode_kernel(const float* __restrict__ deltas,
                       const float* __restrict__ anchors,
                       const int* __restrict__ sel_idx,
                       const float* __restrict__ sel_logit,
                       float4* __restrict__ boxes, float* __restrict__ masked) {
  const int g = blockIdx.x * 256 + threadIdx.x;
  if (g >= B * PRE) return;
  const int b = g / PRE;
  const int i = sel_idx[g];
  const int a = i % A;
  const int hw = i / A;

  // deltas layout: (B, A*4, H, W); channel = a*4 + c
  const size_t base = (((size_t)b * A + a) * 4) * (size_t)HW + (size_t)hw;
  const float dx = deltas[base];
  const float dy = deltas[base + (size_t)HW];
  float dwv      = deltas[base + 2 * (size_t)HW];
  float dhv      = deltas[base + 3 * (size_t)HW];

  const float ax1 = anchors[4 * (size_t)i + 0];
  const float ay1 = anchors[4 * (size_t)i + 1];
  const float ax2 = anchors[4 * (size_t)i + 2];
  const float ay2 = anchors[4 * (size_t)i + 3];

  const float w = ax2 - ax1, h = ay2 - ay1;
  const float cx = ax1 + 0.5f * w, cy = ay1 + 0.5f * h;
  dwv = fminf(dwv, BBOX_CLIP);
  dhv = fminf(dhv, BBOX_CLIP);
  const float pcx = dx * w + cx, pcy = dy * h + cy;
  const float pw = expf(dwv) * w, ph = expf(dhv) * h;

  const float x1 = fminf(fmaxf(pcx - 0.5f * pw, 0.0f), IMGF);
  const float y1 = fminf(fmaxf(pcy - 0.5f * ph, 0.0f), IMGF);
  const float x2 = fminf(fmaxf(pcx + 0.5f * pw, 0.0f), IMGF);
  const float y2 = fminf(fmaxf(pcy + 0.5f * ph, 0.0f), IMGF);

  const float sc = 1.0f / (1.0f + expf(-sel_logit[g]));
  const bool keep = ((x2 - x1) >= 1.0f) && ((y2 - y1) >= 1.0f) && (sc > 0.0f);

  boxes[g] = make_float4(x1, y1, x2, y2);
  masked[g] = keep ? sc : -1.0f;
}

// ============================================================================
// Kernel 3: greedy NMS, 1000 serial steps. One block (1024 thr = 32 waves)
// per image; all state in LDS (~40 KB of the 320 KB WGP pool). Candidate
// staging uses CDNA5 async global->LDS DMA when the toolchain exposes it.
// ============================================================================
__global__ __launch_bounds__(1024)
void rpn_nms_kernel(const float4* __restrict__ boxes_g,
                    const float* __restrict__ masked_g,
                    float* __restrict__ out) {
  const int b = blockIdx.x;
  const int t = threadIdx.x;
  const unsigned lane = threadIdx.x & 31u;
  const unsigned wid  = threadIdx.x >> 5;

  __shared__ float4 lb[PRE];                     // clipped boxes (read-only after load)
  __shared__ float  sw[PRE];                     // masked scores (mutated by suppression)
  __shared__ unsigned long long red[32];
  __shared__ unsigned long long best;

  const float4* __restrict__ bsrc = boxes_g + (size_t)b * PRE;
  const float*  __restrict__ msrc = masked_g + (size_t)b * PRE;

#if defined(RPN_USE_ASYNC_LDS)
  for (int j = t; j < PRE; j += 1024) {
    __builtin_amdgcn_global_load_async_to_lds_b128(
        (gv4p)(bsrc + j), (lv4p)(lb + j), 0, 0);
    __builtin_amdgcn_global_load_async_to_lds_b32(
        (gi1p)(msrc + j), (li1p)(sw + j), 0, 0);
  }
  __builtin_amdgcn_s_wait_asynccnt(0);
#else
  for (int j = t; j < PRE; j += 1024) {
    lb[j] = bsrc[j];
    sw[j] = msrc[j];
  }
#endif
  __syncthreads();

  const int j1 = t + 1024;                       // second element per thread (j1<PRE iff t<976)

  for (int it = 0; it < POST; ++it) {
    // ---- packed argmax: (ordered score)<<32 | ~index  -> first-max tie-break ----
    unsigned long long k0 = ((unsigned long long)fkey(sw[t]) << 32) |
                            (unsigned long long)(0xFFFFFFFFu - (unsigned)t);
    if (j1 < PRE) {
      unsigned long long k1 = ((unsigned long long)fkey(sw[j1]) << 32) |
                              (unsigned long long)(0xFFFFFFFFu - (unsigned)j1);
      if (k1 > k0) k0 = k1;
    }
    for (int off = 16; off > 0; off >>= 1) {
      unsigned long long o = __shfl_down(k0, off);
      if (o > k0) k0 = o;
    }
    if (lane == 0) red[wid] = k0;
    __syncthreads();
    if (wid == 0) {
      unsigned long long v = red[lane];
      for (int off = 16; off > 0; off >>= 1) {
        unsigned long long o = __shfl_down(v, off);
        if (o > v) v = o;
      }
      if (lane == 0) best = v;
    }
    __syncthreads();

    const unsigned long long bk = best;
    const unsigned ord = (unsigned)(bk >> 32);
    const int bi = (int)(0xFFFFFFFFu - (unsigned)(bk & 0xFFFFFFFFu));
    const bool valid = ord > ORD_NEG_HALF;       // score > -0.5
    const float4 pb = lb[bi];

    if (t == 0) {
      float* o = out + ((size_t)b * POST + (size_t)it) * 5;
      if (valid) {
        o[0] = pb.x; o[1] = pb.y; o[2] = pb.z; o[3] = pb.w;
        o[4] = funkey(ord);                      // original sigmoid score
      } else {
        o[0] = 0.0f; o[1] = 0.0f; o[2] = 0.0f; o[3] = 0.0f; o[4] = 0.0f;
      }
    }

    // ---- suppression (self-inclusive, matches reference semantics) ----
    const float areaA = (pb.z - pb.x) * (pb.w - pb.y);
    {
      const float4 bj = lb[t];
      const float ix1 = fmaxf(pb.x, bj.x), iy1 = fmaxf(pb.y, bj.y);
      const float ix2 = fminf(pb.z, bj.z), iy2 = fminf(pb.w, bj.w);
      const float inter = fmaxf(ix2 - ix1, 0.0f) * fmaxf(iy2 - iy1, 0.0f);
      const float areaB = (bj.z - bj.x) * (bj.w - bj.y);
      if (inter / (areaA + areaB - inter + 1e-9f) > NMS_THR) sw[t] = -1.0f;
    }
    if (j1 < PRE) {
      const float4 bj = lb[j1];
      const float ix1 = fmaxf(pb.x, bj.x), iy1 = fmaxf(pb.y, bj.y);
      const float ix2 = fminf(pb.z, bj.z), iy2 = fminf(pb.w, bj.w);
      const float inter = fmaxf(ix2 - ix1, 0.0f) * fmaxf(iy2 - iy1, 0.0f);
      const float areaB = (bj.z - bj.x) * (bj.w - bj.y);
      if (inter / (areaA + areaB - inter + 1e-9f) > NMS_THR) sw[j1] = -1.0f;
    }
    __syncthreads();
  }
}

// ============================================================================
extern "C" void kernel_launch(void* const* d_in, const int* in_sizes, int n_in,
                              void* d_out, int out_size, void* d_ws, size_t ws_size,
                              hipStream_t stream) {
  const float* obj     = (const float*)d_in[0];   // (B, A, H, W)
  const float* deltas  = (const float*)d_in[1];   // (B, A*4, H, W)
  const float* anchors = (const float*)d_in[2];   // (N, 4)
  float* out = (float*)d_out;                     // (B, POST, 5)

  // workspace layout (floats): sel_idx | sel_logit | boxes(float4) | masked
  float* wf = (float*)d_ws;
  int*    sel_idx   = (int*)wf;                                // B*PRE ints
  float*  sel_logit = wf + (size_t)B * PRE;                    // B*PRE
  float4* boxes     = (float4*)(wf + 2 * (size_t)B * PRE);     // B*PRE float4 (16B aligned)
  float*  masked    = wf + 2 * (size_t)B * PRE + 4 * (size_t)B * PRE;  // B*PRE

  rpn_topk_kernel<<<B, 1024, 0, stream>>>(obj, sel_idx, sel_logit);
  rpn_decode_kernel<<<(B * PRE + 255) / 256, 256, 0, stream>>>(
      deltas, anchors, sel_idx, sel_logit, boxes, masked);
  rpn_nms_kernel<<<B, 1024, 0, stream>>>(boxes, masked, out);
}